// GaussianMixture_50379966382117
// MI455X (gfx1250) — compile-verified
//
#include <hip/hip_runtime.h>
#include <math.h>

// ---------------------------------------------------------------------------
// GaussianMixture log-likelihood on MI455X (gfx1250, wave32).
//   y[n] = logsumexp_k( c[k] + a[k]*(x_n . m_k) - hb[k]*||x_n||^2 ) + prior
// with c[k] = pi_term + 16*logbeta[k] + logsoftmax(w)[k] - hb[k]*||m_k||^2,
//      a[k] = exp(logbeta[k]),  hb[k] = 0.5*exp(logbeta[k]).
// Dominant work: X(131072x32) @ M^T(32x64)  -> V_WMMA_F32_16X16X4_F32.
// HBM-bound (~17 MB traffic), so fp32 WMMA keeps reference precision while
// leaving the memory system as the only limiter.
// ---------------------------------------------------------------------------

typedef __attribute__((ext_vector_type(2))) float v2f;
typedef __attribute__((ext_vector_type(8))) float v8f;

#define DIM 32
#define KMIX 64
#define LOG2PI 1.8378770664093453f
#define HALF_LOG2PI 0.9189385332046727f
#define LOGBETA_INIT 1.3862943611198906f   /* -2*log(0.5) */

// ---------------------------------------------------------------------------
// Prep: per-mixture constants + scalar prior.  One block of 256 threads.
// ws layout: c[64] | a[64] | hb[64] | prior[1]
// ---------------------------------------------------------------------------
__global__ void gmm_prep_kernel(const float* __restrict__ mean,
                                const float* __restrict__ logbeta,
                                const float* __restrict__ weight,
                                float* __restrict__ cS,
                                float* __restrict__ aS,
                                float* __restrict__ hbS,
                                float* __restrict__ priorS) {
    __shared__ float red[256];
    __shared__ float lsew_s;
    const int t = threadIdx.x;

    // ---- logsumexp over weights (K=64) ----
    red[t] = (t < KMIX) ? weight[t] : -INFINITY;
    __syncthreads();
    if (t == 0) {
        float mx = -INFINITY;
        for (int k = 0; k < KMIX; ++k) mx = fmaxf(mx, red[k]);
        float s = 0.0f;
        for (int k = 0; k < KMIX; ++k) s += __expf(red[k] - mx);
        lsew_s = mx + __logf(s);
    }
    __syncthreads();
    const float lsew = lsew_s;

    // ---- per-k constants ----
    if (t < KMIX) {
        const float lb = logbeta[t];
        const float hb = 0.5f * __expf(lb);
        float msq = 0.0f;
        for (int d = 0; d < DIM; ++d) {
            float m = mean[t * DIM + d];
            msq += m * m;
        }
        const float pi_term = -0.5f * (float)DIM * LOG2PI;
        cS[t]  = pi_term + 0.5f * (float)DIM * lb + (weight[t] - lsew) - hb * msq;
        aS[t]  = 2.0f * hb;
        hbS[t] = hb;
    }

    // ---- prior: lgamma(64) + sum N(mean;0,1) + sum N(logbeta;INIT,0.5) ----
    float p = 0.0f;
    for (int i = t; i < KMIX * DIM; i += 256) {
        float m = mean[i];
        p += -0.5f * m * m - HALF_LOG2PI;
    }
    if (t < KMIX) {
        float z = (logbeta[t] - LOGBETA_INIT) * 2.0f;   // /0.5
        p += -0.5f * z * z - __logf(0.5f) - HALF_LOG2PI;
    }
    __syncthreads();
    red[t] = p;
    __syncthreads();
    if (t == 0) {
        float s = 0.0f;
        for (int i = 0; i < 256; ++i) s += red[i];
        priorS[0] = s + lgammaf((float)KMIX);
    }
}

// ---------------------------------------------------------------------------
// Main: one wave32 per 16-row tile; 4 WMMA accumulators cover all 64 mixtures.
// ---------------------------------------------------------------------------
#define MEAN_STRIDE 34   // pad 32 -> 34 floats: conflict-free ds_load_b64

__global__ void __launch_bounds__(256)
gmm_main_kernel(const float* __restrict__ x,
                const float* __restrict__ mean,
                const float* __restrict__ cS,
                const float* __restrict__ aS,
                const float* __restrict__ hbS,
                const float* __restrict__ priorS,
                float* __restrict__ out,
                int ntiles) {
    __shared__ float smean[KMIX * MEAN_STRIDE];

    // Stage mean (64x32 f32 = 8 KB) into LDS, coalesced.
    for (int i = threadIdx.x; i < KMIX * DIM; i += blockDim.x) {
        int k = i >> 5, d = i & 31;
        smean[k * MEAN_STRIDE + d] = mean[i];
    }
    __syncthreads();

    const int lane = threadIdx.x & 31;
    const int wv   = threadIdx.x >> 5;
    const int tile = blockIdx.x * 8 + wv;
    if (tile >= ntiles) return;

    const int rowBase = tile * 16;
    const int h   = lane >> 4;   // wave half: selects K pair within a WMMA
    const int l16 = lane & 15;   // row (for A) / column (for B)

    const float* __restrict__ xrow = x + (size_t)(rowBase + l16) * DIM;

    v8f acc0 = {}, acc1 = {}, acc2 = {}, acc3 = {};
    float psq = 0.0f;

    // K-loop: 8 steps of K=4.  A frag: X[row][k0..k0+1] per lane.
    // B frag: mean[n][k0..k0+1] per lane (n = ntile*16 + l16), from LDS.
    #pragma unroll
    for (int ks = 0; ks < 8; ++ks) {
        const int k0 = ks * 4 + 2 * h;
        v2f a;
        a.x = xrow[k0];
        a.y = xrow[k0 + 1];
        psq += a.x * a.x + a.y * a.y;

        v2f b0, b1, b2, b3;
        b0.x = smean[(0 * 16 + l16) * MEAN_STRIDE + k0];
        b0.y = smean[(0 * 16 + l16) * MEAN_STRIDE + k0 + 1];
        b1.x = smean[(1 * 16 + l16) * MEAN_STRIDE + k0];
        b1.y = smean[(1 * 16 + l16) * MEAN_STRIDE + k0 + 1];
        b2.x = smean[(2 * 16 + l16) * MEAN_STRIDE + k0];
        b2.y = smean[(2 * 16 + l16) * MEAN_STRIDE + k0 + 1];
        b3.x = smean[(3 * 16 + l16) * MEAN_STRIDE + k0];
        b3.y = smean[(3 * 16 + l16) * MEAN_STRIDE + k0 + 1];

        acc0 = __builtin_amdgcn_wmma_f32_16x16x4_f32(false, a, false, b0,
                                                     (short)0, acc0, false, false);
        acc1 = __builtin_amdgcn_wmma_f32_16x16x4_f32(false, a, false, b1,
                                                     (short)0, acc1, false, false);
        acc2 = __builtin_amdgcn_wmma_f32_16x16x4_f32(false, a, false, b2,
                                                     (short)0, acc2, false, false);
        acc3 = __builtin_amdgcn_wmma_f32_16x16x4_f32(false, a, false, b3,
                                                     (short)0, acc3, false, false);
    }

    // ||x_row||^2: combine the two wave halves; lane l then holds row (l&15).
    const float xn2v = psq + __shfl_xor(psq, 16, 32);

    // Per-lane mixture columns: n = {0,16,32,48} + l16.
    const float c0 = cS[l16],      c1 = cS[16 + l16];
    const float c2 = cS[32 + l16], c3 = cS[48 + l16];
    const float a0 = aS[l16],      a1 = aS[16 + l16];
    const float a2 = aS[32 + l16], a3 = aS[48 + l16];
    const float q0 = hbS[l16],     q1 = hbS[16 + l16];
    const float q2 = hbS[32 + l16], q3 = hbS[48 + l16];
    const float prior = priorS[0];

    // D layout: VGPR r holds row m = r + 8*h; column = ntile*16 + l16.
    #pragma unroll
    for (int r = 0; r < 8; ++r) {
        const int m = r + 8 * h;
        const float xn2m = __shfl(xn2v, m, 16);

        float t0 = c0 + a0 * acc0[r] - q0 * xn2m;
        float t1 = c1 + a1 * acc1[r] - q1 * xn2m;
        float t2 = c2 + a2 * acc2[r] - q2 * xn2m;
        float t3 = c3 + a3 * acc3[r] - q3 * xn2m;

        float mx = fmaxf(fmaxf(t0, t1), fmaxf(t2, t3));
        #pragma unroll
        for (int s = 1; s < 16; s <<= 1) mx = fmaxf(mx, __shfl_xor(mx, s, 16));

        float sum = __expf(t0 - mx) + __expf(t1 - mx) +
                    __expf(t2 - mx) + __expf(t3 - mx);
        #pragma unroll
        for (int s = 1; s < 16; s <<= 1) sum += __shfl_xor(sum, s, 16);

        if (l16 == r) out[rowBase + m] = mx + __logf(sum) + prior;
    }
}

// ---------------------------------------------------------------------------
extern "C" void kernel_launch(void* const* d_in, const int* in_sizes, int n_in,
                              void* d_out, int out_size, void* d_ws, size_t ws_size,
                              hipStream_t stream) {
    const float* x       = (const float*)d_in[0];
    const float* mean    = (const float*)d_in[1];
    const float* logbeta = (const float*)d_in[2];
    const float* weight  = (const float*)d_in[3];
    float* out = (float*)d_out;

    float* ws     = (float*)d_ws;
    float* cS     = ws;
    float* aS     = ws + KMIX;
    float* hbS    = ws + 2 * KMIX;
    float* priorS = ws + 3 * KMIX;

    const int nrows  = in_sizes[0] / DIM;
    const int ntiles = nrows / 16;            // 131072/16 = 8192
    const int blocks = (ntiles + 7) / 8;      // 8 waves per block

    gmm_prep_kernel<<<1, 256, 0, stream>>>(mean, logbeta, weight,
                                           cS, aS, hbS, priorS);
    gmm_main_kernel<<<blocks, 256, 0, stream>>>(x, mean, cS, aS, hbS, priorS,
                                                out, ntiles);
}